// SSVIModel_54683523613253
// MI455X (gfx1250) — compile-verified
//
#include <hip/hip_runtime.h>

// SSVI power-law surface: out, d(out)/dlogm, d2(out)/dlogm2, plus a zero
// grad_ttm stream. Elementwise over N=2^25 points; memory-bound (24B/elem,
// ~805 MB/call -> ~35us floor at 23.3 TB/s). WMMA has no mapping here;
// optimize the data-movement path: b128 nontemporal loads/stores.
//
// Closed forms (s^2 = pl^2 + 2*rho*pl + 1, pl = phi*logm):
//   out = 0.5*y*(1 + rho*pl + s)
//   g1  = 0.5*y*phi*(rho + (pl+rho)/s)
//   g2  = 0.5*y*phi^2*(1-rho^2)/s^3        [uses s^2-(pl+rho)^2 = 1-rho^2]

typedef float v4f __attribute__((ext_vector_type(4)));

struct SsviR { float o, g1, g2; };

__device__ __forceinline__ SsviR ssvi_point(float lm, float y,
                                            float rho, float eta, float gamma,
                                            float omr2) {
    // phi = eta / (y^gamma * (1+y)^(1-gamma)) = eta * exp(-(g*ln y + (1-g)*ln(1+y)))
    float ly  = logf(y);
    float l1y = log1pf(y);
    float phi = eta * expf(-fmaf(gamma, ly, (1.0f - gamma) * l1y));
    float pl  = phi * lm;
    float t   = fmaf(pl, pl, fmaf(2.0f * rho, pl, 1.0f));   // s^2
    float s   = sqrtf(t);
    float hy  = 0.5f * y;
    float inv_s = 1.0f / s;
    SsviR r;
    r.o  = hy * (fmaf(rho, pl, 1.0f) + s);
    r.g1 = hy * phi * fmaf(pl + rho, inv_s, rho);
    // 1/s^3 = inv_s / t
    r.g2 = hy * phi * phi * omr2 * inv_s / t;
    return r;
}

__global__ __launch_bounds__(256) void ssvi_kernel(
    const float* __restrict__ logm,
    const float* __restrict__ yATM,
    const float* __restrict__ raw_rho,
    const float* __restrict__ raw_eta,
    const float* __restrict__ raw_gamma,
    float* __restrict__ out,   // [4*n]: out | zeros | g1 | g2
    int n)
{
    // Uniform scalar params -> s_load + a few uniform VALU ops (negligible).
    const float rho   = tanhf(raw_rho[0]);
    const float eta   = expf(raw_eta[0]);
    const float gamma = expf(raw_gamma[0]);
    const float omr2  = fmaf(-rho, rho, 1.0f);

    const int n4 = n >> 2;           // float4 chunks
    const int i4 = blockIdx.x * blockDim.x + threadIdx.x;

    if (i4 < n4) {
        const v4f* lm4 = (const v4f*)logm;
        const v4f* yy4 = (const v4f*)yATM;
        v4f lm = __builtin_nontemporal_load(&lm4[i4]);
        v4f yy = __builtin_nontemporal_load(&yy4[i4]);

        v4f o, g1, g2;
#pragma unroll
        for (int k = 0; k < 4; ++k) {
            SsviR r = ssvi_point(lm[k], yy[k], rho, eta, gamma, omr2);
            o[k]  = r.o;
            g1[k] = r.g1;
            g2[k] = r.g2;
        }

        v4f* ov = (v4f*)out;
        const v4f z4 = {0.0f, 0.0f, 0.0f, 0.0f};
        __builtin_nontemporal_store(o,  &ov[i4]);            // output
        __builtin_nontemporal_store(z4, &ov[i4 + n4]);       // grad_ttm1 = 0
        __builtin_nontemporal_store(g1, &ov[i4 + 2 * n4]);   // d/dlogm
        __builtin_nontemporal_store(g2, &ov[i4 + 3 * n4]);   // d2/dlogm2
    }

    // Scalar tail (n % 4 != 0); N=2^25 makes this dead code, kept for safety.
    const int rem = n & 3;
    if (rem && i4 < rem) {
        const int i = (n4 << 2) + i4;
        SsviR r = ssvi_point(logm[i], yATM[i], rho, eta, gamma, omr2);
        out[i]         = r.o;
        out[i + n]     = 0.0f;
        out[i + 2 * n] = r.g1;
        out[i + 3 * n] = r.g2;
    }
}

extern "C" void kernel_launch(void* const* d_in, const int* in_sizes, int n_in,
                              void* d_out, int out_size, void* d_ws, size_t ws_size,
                              hipStream_t stream) {
    (void)n_in; (void)out_size; (void)d_ws; (void)ws_size;
    const float* logm      = (const float*)d_in[0];
    const float* yATM      = (const float*)d_in[1];
    const float* raw_rho   = (const float*)d_in[2];
    const float* raw_eta   = (const float*)d_in[3];
    const float* raw_gamma = (const float*)d_in[4];
    float* out = (float*)d_out;

    const int n  = in_sizes[0];
    const int n4 = n >> 2;
    const int threads = 256;
    const int blocks  = (n4 + threads - 1) / threads;

    ssvi_kernel<<<blocks, threads, 0, stream>>>(logm, yATM, raw_rho, raw_eta,
                                                raw_gamma, out, n);
}